// Simple2DRNN_78219944395388
// MI455X (gfx1250) — compile-verified
//
#include <hip/hip_runtime.h>
#include <hip/hip_bf16.h>

typedef __attribute__((ext_vector_type(2))) float v2f;
typedef __attribute__((ext_vector_type(8))) float v8f;

#define T_STEPS 4096
#define BATCH   2048
#define HID     16

#if __has_builtin(__builtin_amdgcn_tanhf)
__device__ __forceinline__ float fast_tanh(float x) { return __builtin_amdgcn_tanhf(x); }
#else
__device__ __forceinline__ float fast_tanh(float x) { return tanhf(x); }
#endif

__device__ __forceinline__ v8f wmma_k4(v2f a, v2f b, v8f c) {
    // D(16x16,f32) = A(16x4,f32) * B(4x16,f32) + C
    return __builtin_amdgcn_wmma_f32_16x16x4_f32(false, a, false, b, (short)0, c,
                                                 false, false);
}

// State-row permutation: physical D row m holds logical hidden index perm(m).
// Chosen so that B-chunk (c,r) == D VGPR (2c+r) for BOTH lane halves:
//   m = 2q+r  (m<8)  -> logical 4q+r      (lane half 0, contraction K=4c+r)
//   m = 2q+r+8       -> logical 4q+r+2    (lane half 1, contraction K=4c+r+2)
__device__ __forceinline__ int permrow(int m) {
    int mm = m & 7;
    return 4 * (mm >> 1) + (mm & 1) + ((m & 8) ? 2 : 0);
}

__global__ __launch_bounds__(32) void rnn_scan_wmma(
    const float* __restrict__ x_seq,  // (T,B,2)
    const float* __restrict__ h0,     // (1,B,16)
    const float* __restrict__ W_ih,   // (16,2)
    const float* __restrict__ W_hh,   // (16,16)
    const float* __restrict__ b_ih,   // (16)
    const float* __restrict__ b_hh,   // (16)
    const float* __restrict__ W_fc,   // (2,16)
    const float* __restrict__ b_fc,   // (2)
    float* __restrict__ y_out,        // (T,B,2)
    float* __restrict__ h_last)       // (1,B,16)
{
    const int lane = threadIdx.x;
    const int n    = lane & 15;   // N / batch column within tile
    const int hi   = lane >> 4;   // lane half
    const bool lo16 = (hi == 0);
    const int b0   = blockIdx.x * 16;

    // ---- loop-invariant weights in A-operand layout (rows pre-permuted) ----
    // A-layout (chunk c, reg r): lane L holds [M=L%16][K = 4c + r + 2*(L/16)]
    const int pm = permrow(n);    // logical row for this lane's physical M

    v2f AW[4];                    // W_hh: out rows permuted, K logical
#pragma unroll
    for (int c = 0; c < 4; ++c) {
        AW[c][0] = W_hh[pm * 16 + 4 * c + 0 + 2 * hi];
        AW[c][1] = W_hh[pm * 16 + 4 * c + 1 + 2 * hi];
    }
    // W_fc: rows 0,1 real; rows 8,9 duplicate rows 0,1 so lanes 16-31 carry
    // bit-identical y values (enables an unconditional full-wave y store).
    v2f AF[4];
#pragma unroll
    for (int c = 0; c < 4; ++c) {
        bool valid = (n & 7) < 2;
        int  orow  = n & 7;
        AF[c][0] = valid ? W_fc[orow * 16 + 4 * c + 0 + 2 * hi] : 0.0f;
        AF[c][1] = valid ? W_fc[orow * 16 + 4 * c + 1 + 2 * hi] : 0.0f;
    }
    v2f AI;                       // W_ih (16x2) padded to K=4, rows permuted
    AI[0] = lo16 ? W_ih[pm * 2 + 0] : 0.0f;
    AI[1] = lo16 ? W_ih[pm * 2 + 1] : 0.0f;

    // bias in C/D layout of pre^T (rows permuted)
    v8f biasC;
#pragma unroll
    for (int v = 0; v < 8; ++v) {
        int m = permrow(v + 8 * hi);
        biasC[v] = b_ih[m] + b_hh[m];
    }
    const float bfc0 = b_fc[0];
    const float bfc1 = b_fc[1];
    const v8f vzero = {};

    // ---- initial state h0^T, permuted rows, directly as B chunks ----
    v2f hB[4];
#pragma unroll
    for (int c = 0; c < 4; ++c) {
        hB[c][0] = h0[(size_t)(b0 + n) * 16 + permrow(2 * c + 0 + 8 * hi)];
        hB[c][1] = h0[(size_t)(b0 + n) * 16 + permrow(2 * c + 1 + 8 * hi)];
    }

    // x_t^T in B-layout: rows 0,1 = x (lanes 0-15), rows 2,3 zero.
    // All lanes load (upper half duplicates partner address), then mask.
    const float* xrow = x_seq + (size_t)(b0 + n) * 2;
    v2f xv = __builtin_nontemporal_load((const v2f*)(xrow));
    v2f xcur;
    xcur[0] = lo16 ? xv[0] : 0.0f;
    xcur[1] = lo16 ? xv[1] : 0.0f;

    v8f preCur = wmma_k4(AI, xcur, biasC);   // pre^T for t=0, off the chain

    for (int t = 0; t < T_STEPS; ++t) {
        // prefetch x_{t+1} (clamped, branch-free)
        int tn = (t + 1 < T_STEPS) ? (t + 1) : t;
        v2f xw = __builtin_nontemporal_load(
            (const v2f*)(xrow + (size_t)tn * BATCH * 2));
        v2f xn;
        xn[0] = lo16 ? xw[0] : 0.0f;
        xn[1] = lo16 ? xw[1] : 0.0f;

        // recurrence: 4 independent WMMAs (B chunks are hB[] directly), tree add
        v8f a0 = wmma_k4(AW[0], hB[0], preCur);
        v8f a1 = wmma_k4(AW[1], hB[1], vzero);
        v8f a2 = wmma_k4(AW[2], hB[2], vzero);
        v8f a3 = wmma_k4(AW[3], hB[3], vzero);

        v8f preN = wmma_k4(AI, xn, biasC);   // next step's pre^T, off the chain

        v8f s;
#pragma unroll
        for (int v = 0; v < 8; ++v) s[v] = (a0[v] + a1[v]) + (a2[v] + a3[v]);

        // h_t = tanh(s): D VGPR (2c+r) IS next B-chunk (c,r) — no shuffles
#pragma unroll
        for (int c = 0; c < 4; ++c) {
            hB[c][0] = fast_tanh(s[2 * c + 0]);
            hB[c][1] = fast_tanh(s[2 * c + 1]);
        }

        // y_t^T = W_fc_pad * h_t^T (rows 0,1 valid; 8,9 are duplicates)
        v8f y = wmma_k4(AF[0], hB[0], vzero);
        y     = wmma_k4(AF[1], hB[1], y);
        y     = wmma_k4(AF[2], hB[2], y);
        y     = wmma_k4(AF[3], hB[3], y);

        // full-wave store: lanes L and L+16 write identical data to same addr
        v2f yo;
        yo[0] = y[0] + bfc0;
        yo[1] = y[1] + bfc1;
        __builtin_nontemporal_store(
            yo, (v2f*)(y_out + ((size_t)t * BATCH + b0 + n) * 2));

        preCur = preN;
    }

    // final state (un-permute rows)
#pragma unroll
    for (int c = 0; c < 4; ++c) {
        h_last[(size_t)(b0 + n) * 16 + permrow(2 * c + 0 + 8 * hi)] = hB[c][0];
        h_last[(size_t)(b0 + n) * 16 + permrow(2 * c + 1 + 8 * hi)] = hB[c][1];
    }
}

extern "C" void kernel_launch(void* const* d_in, const int* in_sizes, int n_in,
                              void* d_out, int out_size, void* d_ws, size_t ws_size,
                              hipStream_t stream) {
    (void)in_sizes; (void)n_in; (void)out_size; (void)d_ws; (void)ws_size;
    const float* x_seq = (const float*)d_in[0];
    const float* h0    = (const float*)d_in[1];
    const float* W_ih  = (const float*)d_in[2];
    const float* W_hh  = (const float*)d_in[3];
    const float* b_ih  = (const float*)d_in[4];
    const float* b_hh  = (const float*)d_in[5];
    const float* W_fc  = (const float*)d_in[6];
    const float* b_fc  = (const float*)d_in[7];

    float* y_out  = (float*)d_out;                       // T*B*2 floats
    float* h_last = y_out + (size_t)T_STEPS * BATCH * 2; // B*16 floats

    dim3 grid(BATCH / 16);   // 128 tiles, one wave each
    dim3 block(32);
    rnn_scan_wmma<<<grid, block, 0, stream>>>(x_seq, h0, W_ih, W_hh, b_ih, b_hh,
                                              W_fc, b_fc, y_out, h_last);
}